// DBSCAN_41644002902456
// MI455X (gfx1250) — compile-verified
//
#include <hip/hip_runtime.h>

typedef __attribute__((ext_vector_type(2))) float v2f;
typedef __attribute__((ext_vector_type(4))) float v4f;
typedef __attribute__((ext_vector_type(8))) float v8f;

#define N_PTS   8192
#define DIMS    16
#define EPS2    0.25f
#define MIN_PTS 10
#define BATCH   4

// ---------------------------------------------------------------------------
// Pass 1: sq[b*N + n] = ||x[b][n]||^2   (32768 values, 128KB in d_ws)
// ---------------------------------------------------------------------------
__global__ __launch_bounds__(256) void dbscan_sqnorm(
    const float* __restrict__ x, float* __restrict__ sq) {
  int idx = blockIdx.x * 256 + threadIdx.x;          // one point per thread
  const v4f* p = (const v4f*)(x + (size_t)idx * DIMS); // 64B-aligned rows
  float s = 0.f;
#pragma unroll
  for (int q = 0; q < 4; ++q) {
    v4f v = p[q];
    s += v.x * v.x + v.y * v.y + v.z * v.z + v.w * v.w;
  }
  sq[idx] = s;
}

// ---------------------------------------------------------------------------
// Pass 2: one wave32 owns a 32-row i-slab (two 16-row WMMA tiles) and sweeps
// all 512 j-tiles. Gram tiles via chained V_WMMA_F32_16X16X4_F32; B tile and
// sq_j load are shared across both accumulator chains.
// ---------------------------------------------------------------------------
__global__ __launch_bounds__(256) void dbscan_wmma_f32(
    const float* __restrict__ x, const float* __restrict__ sqw,
    int* __restrict__ out) {
  const int lane      = threadIdx.x & 31;
  const int waveInBlk = threadIdx.x >> 5;
  const int waveId    = blockIdx.x * 8 + waveInBlk;   // 1024 waves total
  const int b         = waveId >> 8;                  // 256 slabs per batch
  const int i0        = (waveId & 255) << 5;          // 32-row slab
  const float*  xb    = x   + (size_t)b * N_PTS * DIMS;
  const float*  sqb   = sqw + (size_t)b * N_PTS;

  const int m  = lane & 15;   // tile row (A) / tile col (B)
  const int hi = lane >> 4;   // half-wave -> K sub-pair

  // ---- Preload two A tiles (rows i0..i0+15, i0+16..i0+31) ----
  v2f a0[4], a1[4];
  const float* ar0 = xb + (i0 + m) * DIMS + 2 * hi;
  const float* ar1 = ar0 + 16 * DIMS;
#pragma unroll
  for (int kc = 0; kc < 4; ++kc) {
    a0[kc] = *(const v2f*)(ar0 + 4 * kc);
    a1[kc] = *(const v2f*)(ar1 + 4 * kc);
  }

  // Per-output-VGPR thresholds from precomputed norms (C VGPR r -> M = r+8*hi)
  float thr0[8], thr1[8];
#pragma unroll
  for (int r = 0; r < 8; ++r) {
    thr0[r] = EPS2 - sqb[i0 + r + 8 * hi];
    thr1[r] = EPS2 - sqb[i0 + 16 + r + 8 * hi];
  }

  int cnt0[8], cnt1[8];
#pragma unroll
  for (int r = 0; r < 8; ++r) { cnt0[r] = 0; cnt1[r] = 0; }

  // ---- Sweep j-tiles ----
  for (int j0 = 0; j0 < N_PTS; j0 += 16) {
    v2f bm[4];
    const float* brow = xb + (j0 + m) * DIMS + 2 * hi;
#pragma unroll
    for (int kc = 0; kc < 4; ++kc)
      bm[kc] = *(const v2f*)(brow + 4 * kc);

    float sq_j = sqb[j0 + m];          // precomputed ||x_j||^2, L2 hit

    v8f c0 = {}, c1 = {};
#pragma unroll
    for (int kc = 0; kc < 4; ++kc) {   // interleave independent chains
      c0 = __builtin_amdgcn_wmma_f32_16x16x4_f32(
          false, a0[kc], false, bm[kc], (short)0, c0, false, false);
      c1 = __builtin_amdgcn_wmma_f32_16x16x4_f32(
          false, a1[kc], false, bm[kc], (short)0, c1, false, false);
    }

    // d2 < eps2  <=>  sq_j - 2G < (eps2 - sq_i)
#pragma unroll
    for (int r = 0; r < 8; ++r) {
      cnt0[r] += (sq_j - 2.0f * c0[r] < thr0[r]) ? 1 : 0;
      cnt1[r] += (sq_j - 2.0f * c1[r] < thr1[r]) ? 1 : 0;
    }
  }

  // ---- Reduce row counts across each 16-lane half, emit labels ----
  int* ob = out + (size_t)b * N_PTS;
#pragma unroll
  for (int r = 0; r < 8; ++r) {
    int t0 = cnt0[r], t1 = cnt1[r];
    t0 += __shfl_xor(t0, 8, 16);  t1 += __shfl_xor(t1, 8, 16);
    t0 += __shfl_xor(t0, 4, 16);  t1 += __shfl_xor(t1, 4, 16);
    t0 += __shfl_xor(t0, 2, 16);  t1 += __shfl_xor(t1, 2, 16);
    t0 += __shfl_xor(t0, 1, 16);  t1 += __shfl_xor(t1, 1, 16);
    if (m == 0) {
      ob[i0 + r + 8 * hi]      = (t0 < MIN_PTS) ? -1 : 0;
      ob[i0 + 16 + r + 8 * hi] = (t1 < MIN_PTS) ? -1 : 0;
    }
  }
}

extern "C" void kernel_launch(void* const* d_in, const int* in_sizes, int n_in,
                              void* d_out, int out_size, void* d_ws, size_t ws_size,
                              hipStream_t stream) {
  (void)in_sizes; (void)n_in; (void)out_size; (void)ws_size;
  const float* x  = (const float*)d_in[0];
  float* sq       = (float*)d_ws;        // 4*8192 floats = 128KB scratch
  int* out        = (int*)d_out;         // reference output dtype is int32

  // Pass 1: squared norms (32768 threads)
  hipLaunchKernelGGL(dbscan_sqnorm, dim3((BATCH * N_PTS) / 256), dim3(256),
                     0, stream, x, sq);
  // Pass 2: 4 batches * 256 slabs = 1024 waves; 8 waves (256 thr) per block.
  hipLaunchKernelGGL(dbscan_wmma_f32, dim3(128), dim3(256),
                     0, stream, x, sq, out);
}